// TorchVQCRouter_22101901705752
// MI455X (gfx1250) — compile-verified
//
#include <hip/hip_runtime.h>
#include <math.h>

typedef __attribute__((ext_vector_type(2))) float v2f;
typedef __attribute__((ext_vector_type(8))) float v8f;
typedef __attribute__((ext_vector_type(4))) int   v4i;

#define VQC_AS1 __attribute__((address_space(1)))
#define VQC_AS3 __attribute__((address_space(3)))

// ---- CDNA5 async global->LDS path (guarded; fallback = sync staging) --------
#if defined(__has_builtin)
#if __has_builtin(__builtin_amdgcn_global_load_async_to_lds_b128) && \
    __has_builtin(__builtin_amdgcn_s_wait_asynccnt)
#define VQC_ASYNC_LDS 1
#endif
#endif
#ifndef VQC_ASYNC_LDS
#define VQC_ASYNC_LDS 0
#endif

namespace {
constexpr int   kQubits   = 6;
constexpr int   kLayers   = 6;
constexpr int   kClasses  = 35;
constexpr int   kInputDim = 768;
constexpr int   kDim      = 64;            // 2^6 amplitudes
constexpr int   kRows     = 128;           // batch rows per workgroup
constexpr int   kThreads  = 128;           // 4 waves of 32
constexpr int   kKC       = 32;            // K-chunk staged in LDS
constexpr int   kChunks   = kInputDim / kKC;
constexpr int   kFStr     = kKC + 4;       // LDS row stride (floats), bank-spread
constexpr int   kOStr     = kClasses + 1;  // padded output staging stride
constexpr float kPi       = 3.14159265358979323846f;
}

// ---------------- quantum gate helpers (compile-time strides => registers) ---

template<int BIT>
__device__ __forceinline__ void apply_ry(v2f* st, float c, float s) {
#pragma unroll
  for (int i = 0; i < kDim; ++i) {
    if ((i & (1 << BIT)) == 0) {
      const int j = i | (1 << BIT);
      const v2f a0 = st[i], a1 = st[j];
      st[i] = c * a0 - s * a1;
      st[j] = s * a0 + c * a1;
    }
  }
}

template<int BIT>
__device__ __forceinline__ void apply_rot(v2f* st, const float* g) {
  const float g00r = g[0], g00i = g[1], g01r = g[2], g01i = g[3];
  const float g10r = g[4], g10i = g[5], g11r = g[6], g11i = g[7];
#pragma unroll
  for (int i = 0; i < kDim; ++i) {
    if ((i & (1 << BIT)) == 0) {
      const int j = i | (1 << BIT);
      const v2f a0 = st[i], a1 = st[j];
      v2f a0n, a1n;
      a0n.x = -a0.y; a0n.y = a0.x;   // i * a0
      a1n.x = -a1.y; a1n.y = a1.x;   // i * a1
      // complex 2x2 matvec expressed as packed-f32 FMAs
      st[i] = g00r * a0 + g00i * a0n + g01r * a1 + g01i * a1n;
      st[j] = g10r * a0 + g10i * a0n + g11r * a1 + g11i * a1n;
    }
  }
}

template<int CBIT, int TBIT>
__device__ __forceinline__ void apply_cnot(v2f* st) {
#pragma unroll
  for (int i = 0; i < kDim; ++i) {
    if (((i >> CBIT) & 1) && !((i >> TBIT) & 1)) {
      const int j = i | (1 << TBIT);
      const v2f t = st[i]; st[i] = st[j]; st[j] = t;
    }
  }
}

// ---------------- fused kernel ----------------------------------------------

__global__ __launch_bounds__(kThreads)
void vqc_router_fused(const float* __restrict__ features,
                      const float* __restrict__ proj_w,
                      const float* __restrict__ proj_b,
                      const float* __restrict__ vqc_w,
                      const float* __restrict__ lin_w,
                      const float* __restrict__ lin_b,
                      float* __restrict__ out) {
  __shared__ __align__(16) float ldsW[kQubits * kInputDim];    // 18 KB proj_w
  __shared__ __align__(16) float ldsF0[kRows * kFStr];         // 18 KB tile buf 0
  __shared__ __align__(16) float ldsF1[kRows * kFStr];         // 18 KB tile buf 1
  __shared__ __align__(16) float ldsG[kLayers * kQubits * 8];  // Rot gate matrices
  __shared__ __align__(16) float ldsZ[kRows * 8];              // projected angles
  __shared__ __align__(16) float ldsLW[kQubits * kClasses];
  __shared__ __align__(16) float ldsLB[kClasses];
  __shared__ __align__(16) float ldsO[kRows * kOStr];          // 18 KB out staging

  const int tid  = threadIdx.x;
  const int lane = tid & 31;
  const int wv   = tid >> 5;     // wave id 0..3
  const int m    = lane & 15;    // row within 16x16 tile / B column
  const int kh   = lane >> 4;    // half-wave selects K pair {0,1} vs {2,3}

  // ---- stage small operands --------------------------------------------------
  for (int i = tid; i < kQubits * kInputDim; i += kThreads) ldsW[i] = proj_w[i];
  for (int i = tid; i < kQubits * kClasses;  i += kThreads) ldsLW[i] = lin_w[i];
  if (tid < kClasses) ldsLB[tid] = lin_b[tid];

  // Rot(phi,theta,omega) = RZ(omega) RY(theta) RZ(phi); one thread per gate
  if (tid < kLayers * kQubits) {
    const float phi = vqc_w[tid * 3 + 0];
    const float th  = vqc_w[tid * 3 + 1];
    const float om  = vqc_w[tid * 3 + 2];
    float sth, cth, sap, cap, sam, cam;
    sincosf(0.5f * th, &sth, &cth);
    sincosf(0.5f * (phi + om), &sap, &cap);
    sincosf(0.5f * (phi - om), &sam, &cam);
    float* g = &ldsG[tid * 8];
    g[0] =  cap * cth;  g[1] = -sap * cth;   // m00 = e^{-i(phi+om)/2} c
    g[2] = -cam * sth;  g[3] = -sam * sth;   // m01 = -e^{+i(phi-om)/2} s
    g[4] =  cam * sth;  g[5] = -sam * sth;   // m10 =  e^{-i(phi-om)/2} s
    g[6] =  cap * cth;  g[7] =  sap * cth;   // m11 = e^{+i(phi+om)/2} c
  }

  // ---- projection GEMM: x = features @ proj_w^T, via V_WMMA_F32_16X16X4_F32 -
  const float bmask = (m < kQubits) ? 1.0f : 0.0f;   // pad N=6 -> 16 with zeros
  const int   nsafe = (m < kQubits) ? m : 0;
  const float pb    = (m < kQubits) ? proj_b[m] : 0.0f;

  v8f acc0, acc1;
#pragma unroll
  for (int v = 0; v < 8; ++v) { acc0[v] = 0.0f; acc1[v] = 0.0f; }

  const float* frow = features + (size_t)blockIdx.x * kRows * kInputDim;
  const int rbase0 = wv * 32 + m;        // wave's first 16-row tile
  const int rbase1 = wv * 32 + 16 + m;   // wave's second 16-row tile

  // 16 wmma per chunk: two 16-row tiles per wave, 8 K-steps of 4
  auto compute_chunk = [&](int k0, const float* buf) {
#pragma unroll
    for (int kk = 0; kk < kKC; kk += 4) {
      const int kl = kk + 2 * kh;  // this half-wave's K pair within the step
      v2f b;
      b.x = ldsW[nsafe * kInputDim + k0 + kl]     * bmask;
      b.y = ldsW[nsafe * kInputDim + k0 + kl + 1] * bmask;
      const v2f a0 = *reinterpret_cast<const v2f*>(&buf[rbase0 * kFStr + kl]);
      acc0 = __builtin_amdgcn_wmma_f32_16x16x4_f32(false, a0, false, b,
                                                   (short)0, acc0, false, false);
      const v2f a1 = *reinterpret_cast<const v2f*>(&buf[rbase1 * kFStr + kl]);
      acc1 = __builtin_amdgcn_wmma_f32_16x16x4_f32(false, a1, false, b,
                                                   (short)0, acc1, false, false);
    }
  };

#if VQC_ASYNC_LDS
  // ---- double-buffered async global->LDS pipeline (ASYNCcnt) ----------------
  auto issue_chunk = [&](int k0, float* buf) {
#pragma unroll
    for (int it = 0; it < (kRows * (kKC / 4)) / kThreads; ++it) {
      const int i  = tid + it * kThreads;
      const int r  = i >> 3;        // / (kKC/4)
      const int c4 = i & 7;
      __builtin_amdgcn_global_load_async_to_lds_b128(
          (VQC_AS1 v4i*)(frow + (size_t)r * kInputDim + k0 + c4 * 4),
          (VQC_AS3 v4i*)&buf[r * kFStr + c4 * 4],
          0, 0);
    }
  };

  issue_chunk(0, ldsF0);
  __builtin_amdgcn_s_wait_asynccnt(0);
  __syncthreads();   // also covers ldsW/ldsG/ldsLW staging above
  for (int c = 0; c < kChunks; ++c) {
    const float* cur = (c & 1) ? ldsF1 : ldsF0;
    if (c + 1 < kChunks) issue_chunk((c + 1) * kKC, (c & 1) ? ldsF0 : ldsF1);
    compute_chunk(c * kKC, cur);
    if (c + 1 < kChunks) {
      __builtin_amdgcn_s_wait_asynccnt(0);
      __syncthreads();
    }
  }
#else
  // ---- fallback: synchronous VGPR-staged tiles ------------------------------
  for (int k0 = 0; k0 < kInputDim; k0 += kKC) {
    __syncthreads();
#pragma unroll
    for (int it = 0; it < (kRows * (kKC / 4)) / kThreads; ++it) {
      const int i  = tid + it * kThreads;
      const int r  = i >> 3;
      const int c4 = i & 7;
      const float4 v =
          *reinterpret_cast<const float4*>(frow + (size_t)r * kInputDim + k0 + c4 * 4);
      *reinterpret_cast<float4*>(&ldsF0[r * kFStr + c4 * 4]) = v;
    }
    if (k0 + kKC < kInputDim) {
      __builtin_prefetch(frow + (size_t)tid * kInputDim + k0 + kKC, 0, 0);
    }
    __syncthreads();
    compute_chunk(k0, ldsF0);
  }
#endif

  // C layout: lane holds N=m, VGPR v holds row M = v + 8*kh
  if (m < kQubits) {
#pragma unroll
    for (int v = 0; v < 8; ++v) {
      ldsZ[(wv * 32 + 8 * kh + v) * 8 + m]      = acc0[v] + pb;
      ldsZ[(wv * 32 + 16 + 8 * kh + v) * 8 + m] = acc1[v] + pb;
    }
  }
  __syncthreads();

  // ---- 6-qubit state-vector sim, one batch element per thread ----------------
  v2f st[kDim];
  v2f zero; zero.x = 0.0f; zero.y = 0.0f;
#pragma unroll
  for (int i = 0; i < kDim; ++i) st[i] = zero;
  st[0].x = 1.0f;

  {
    float s, c;
    sincosf(0.5f * kPi * tanhf(ldsZ[tid * 8 + 0]), &s, &c); apply_ry<5>(st, c, s);
    sincosf(0.5f * kPi * tanhf(ldsZ[tid * 8 + 1]), &s, &c); apply_ry<4>(st, c, s);
    sincosf(0.5f * kPi * tanhf(ldsZ[tid * 8 + 2]), &s, &c); apply_ry<3>(st, c, s);
    sincosf(0.5f * kPi * tanhf(ldsZ[tid * 8 + 3]), &s, &c); apply_ry<2>(st, c, s);
    sincosf(0.5f * kPi * tanhf(ldsZ[tid * 8 + 4]), &s, &c); apply_ry<1>(st, c, s);
    sincosf(0.5f * kPi * tanhf(ldsZ[tid * 8 + 5]), &s, &c); apply_ry<0>(st, c, s);
  }

#pragma unroll 1
  for (int l = 0; l < kLayers; ++l) {
    const float* g = &ldsG[l * kQubits * 8];
    apply_rot<5>(st, g +  0);
    apply_rot<4>(st, g +  8);
    apply_rot<3>(st, g + 16);
    apply_rot<2>(st, g + 24);
    apply_rot<1>(st, g + 32);
    apply_rot<0>(st, g + 40);
    apply_cnot<5, 4>(st);   // CNOT(0,1)
    apply_cnot<4, 3>(st);   // CNOT(1,2)
    apply_cnot<3, 2>(st);   // CNOT(2,3)
    apply_cnot<2, 1>(st);   // CNOT(3,4)
    apply_cnot<1, 0>(st);   // CNOT(4,5)
    apply_cnot<0, 5>(st);   // CNOT(5,0)
  }

  // PauliZ expectations; qubit q <-> bit (5-q)
  float z[kQubits];
#pragma unroll
  for (int q = 0; q < kQubits; ++q) z[q] = 0.0f;
#pragma unroll
  for (int i = 0; i < kDim; ++i) {
    const float p = st[i].x * st[i].x + st[i].y * st[i].y;
    z[0] += (i & 32) ? -p : p;
    z[1] += (i & 16) ? -p : p;
    z[2] += (i &  8) ? -p : p;
    z[3] += (i &  4) ? -p : p;
    z[4] += (i &  2) ? -p : p;
    z[5] += (i &  1) ? -p : p;
  }

  // ---- final linear 6 -> 35, staged in LDS for coalesced global stores -------
#pragma unroll 1
  for (int c = 0; c < kClasses; ++c) {
    float acc = ldsLB[c];
#pragma unroll
    for (int q = 0; q < kQubits; ++q) acc += z[q] * ldsLW[q * kClasses + c];
    ldsO[tid * kOStr + c] = acc;
  }
  __syncthreads();

  float* op = out + (size_t)blockIdx.x * kRows * kClasses;
  for (int i = tid; i < kRows * kClasses; i += kThreads) {
    const int r = i / kClasses;
    const int c = i - r * kClasses;
    op[i] = ldsO[r * kOStr + c];
  }
}

// ---------------- host launch ------------------------------------------------

extern "C" void kernel_launch(void* const* d_in, const int* in_sizes, int n_in,
                              void* d_out, int out_size, void* d_ws, size_t ws_size,
                              hipStream_t stream) {
  (void)n_in; (void)out_size; (void)d_ws; (void)ws_size;
  const float* features = (const float*)d_in[0];
  const float* proj_w   = (const float*)d_in[1];
  const float* proj_b   = (const float*)d_in[2];
  const float* vqc_w    = (const float*)d_in[3];
  const float* lin_w    = (const float*)d_in[4];
  const float* lin_b    = (const float*)d_in[5];
  float* out = (float*)d_out;

  const int batch = in_sizes[0] / kInputDim;   // 32768
  const int grid  = batch / kRows;             // 256 workgroups
  hipLaunchKernelGGL(vqc_router_fused, dim3(grid), dim3(kThreads), 0, stream,
                     features, proj_w, proj_b, vqc_w, lin_w, lin_b, out);
}